// LinkLabelPredModel_7765300871785
// MI455X (gfx1250) — compile-verified
//
#include <hip/hip_runtime.h>

#define NA  100000
#define NH  20000
#define NE  1000000
#define ELN 200000
#define DD  128

typedef float v2f __attribute__((ext_vector_type(2)));
typedef float v8f __attribute__((ext_vector_type(8)));

// ---------------- degree count: one thread per edge, hw fp32 atomic ----------------
__global__ void deg_count_kernel(const int* __restrict__ dst, float* __restrict__ deg, int nE) {
    int e = blockIdx.x * blockDim.x + threadIdx.x;
    if (e < nE) unsafeAtomicAdd(&deg[dst[e]], 1.0f);
}

// deg -> 1/max(deg,1) in place
__global__ void rdeg_kernel(float* __restrict__ deg, int n) {
    int i = blockIdx.x * blockDim.x + threadIdx.x;
    if (i < n) deg[i] = 1.0f / fmaxf(deg[i], 1.0f);
}

// ---------------- scatter-add of 128-f32 rows: one wave per edge ----------------
// float4 (b128) gather of the source row, 4 global_atomic_add_f32 per lane.
__global__ void scatter_kernel(const float* __restrict__ xsrc,
                               const int* __restrict__ src,
                               const int* __restrict__ dst,
                               float* __restrict__ agg, int nE) {
    int wv   = threadIdx.x >> 5;
    int lane = threadIdx.x & 31;
    int e = blockIdx.x * (blockDim.x >> 5) + wv;
    if (e >= nE) return;
    int s = src[e], d = dst[e];
    const float4 v = ((const float4*)(xsrc + (size_t)s * DD))[lane];
    float* out = agg + (size_t)d * DD + lane * 4;
    unsafeAtomicAdd(out + 0, v.x);
    unsafeAtomicAdd(out + 1, v.y);
    unsafeAtomicAdd(out + 2, v.z);
    unsafeAtomicAdd(out + 3, v.w);
}

// ---------------- fused SAGE transform via V_WMMA_F32_16X16X4_F32 ----------------
// out[nrows,128] = (A1 * rdeg_row) @ W1[K1,128] + bias + (A2 @ W2[128,128]) ; optional relu
// Block = 256 threads = 8 waves; wave w owns output columns [16w,16w+16); block owns 16 rows.
// A 16x4 f32 fragment: lane l holds A[l&15][k0+2*(l>>4)] (+1); B mirrors with N on lanes.
// C/D 16x16 f32: VGPR r -> M = r + 8*(lane>>4), N = lane&15.
__global__ __launch_bounds__(256) void sage_gemm_kernel(
        const float* __restrict__ A1, const float* __restrict__ rdeg,
        const float* __restrict__ W1, const float* __restrict__ bias,
        const float* __restrict__ A2, const float* __restrict__ W2,
        float* __restrict__ out, int K1, int do_relu) {
    const int lane = threadIdx.x & 31;
    const int wv   = threadIdx.x >> 5;      // column tile 0..7
    const int hi   = lane >> 4;             // 0/1 (K half for A/B fragments)
    const int m    = lane & 15;
    const size_t row = (size_t)blockIdx.x * 16 + m;   // A row owned by this lane
    const int    col = wv * 16 + m;                   // B/D column owned by this lane

    v8f acc = {};
    const float scale = rdeg ? rdeg[row] : 1.0f;      // mean-agg normalization folded in
    const float* a1 = A1 + row * (size_t)K1;

    for (int k0 = 0; k0 < K1; k0 += 4) {
        const int ka = k0 + 2 * hi;
        v2f a, b;
        a.x = a1[ka]     * scale;
        a.y = a1[ka + 1] * scale;
        b.x = W1[(size_t)ka      * DD + col];
        b.y = W1[(size_t)(ka + 1) * DD + col];
        acc = __builtin_amdgcn_wmma_f32_16x16x4_f32(false, a, false, b,
                                                    (short)0, acc, false, false);
    }
    if (A2) {   // root/self term: x_dst @ W_r  (K = 128)
        const float* a2 = A2 + row * (size_t)DD;
        for (int k0 = 0; k0 < DD; k0 += 4) {
            const int ka = k0 + 2 * hi;
            v2f a, b;
            a.x = a2[ka];
            a.y = a2[ka + 1];
            b.x = W2[(size_t)ka      * DD + col];
            b.y = W2[(size_t)(ka + 1) * DD + col];
            acc = __builtin_amdgcn_wmma_f32_16x16x4_f32(false, a, false, b,
                                                        (short)0, acc, false, false);
        }
    }
    const float bv = bias[col];
    #pragma unroll
    for (int r = 0; r < 8; ++r) {
        const size_t orow = (size_t)blockIdx.x * 16 + r + 8 * hi;
        float v = acc[r] + bv;
        if (do_relu) v = fmaxf(v, 0.0f);
        out[orow * DD + col] = v;
    }
}

// ---------------- gather-concat z1 = [z_a[row] | z_h[col]] : one wave per link ----------------
__global__ void concat_kernel(const float* __restrict__ za, const float* __restrict__ zh,
                              const int* __restrict__ row, const int* __restrict__ col,
                              float* __restrict__ z1, int n) {
    int wv   = threadIdx.x >> 5;
    int lane = threadIdx.x & 31;
    int e = blockIdx.x * (blockDim.x >> 5) + wv;
    if (e >= n) return;
    const float4* pa = (const float4*)(za + (size_t)row[e] * DD);
    const float4* ph = (const float4*)(zh + (size_t)col[e] * DD);
    float4* o = (float4*)(z1 + (size_t)e * 2 * DD);
    o[lane]      = pa[lane];   // first 128 floats
    o[lane + 32] = ph[lane];   // second 128 floats
}

// ---------------- final projection: pred[e] = z2[e] . W_lin2 + b ----------------
__global__ void pred_kernel(const float* __restrict__ z2, const float* __restrict__ w,
                            const float* __restrict__ b, float* __restrict__ pred, int n) {
    int e = blockIdx.x * blockDim.x + threadIdx.x;
    if (e >= n) return;
    const float4* z  = (const float4*)(z2 + (size_t)e * DD);
    const float4* wp = (const float4*)w;
    float s = 0.0f;
    #pragma unroll 8
    for (int i = 0; i < 32; ++i) {
        float4 a = z[i], c = wp[i];
        s += a.x * c.x + a.y * c.y + a.z * c.z + a.w * c.w;
    }
    pred[e] = s + b[0];
}

extern "C" void kernel_launch(void* const* d_in, const int* in_sizes, int n_in,
                              void* d_out, int out_size, void* d_ws, size_t ws_size,
                              hipStream_t stream) {
    const float* x_author = (const float*)d_in[0];
    const float* x_hotel  = (const float*)d_in[1];
    const int* ah_src  = (const int*)d_in[2];
    const int* ah_dst  = (const int*)d_in[3];
    const int* ha_src  = (const int*)d_in[4];
    const int* ha_dst  = (const int*)d_in[5];
    const int* eli_row = (const int*)d_in[6];
    const int* eli_col = (const int*)d_in[7];
    const float* W1_ah_l = (const float*)d_in[8];
    const float* W1_ah_r = (const float*)d_in[9];
    const float* W1_ha_l = (const float*)d_in[10];
    const float* W1_ha_r = (const float*)d_in[11];
    const float* W2_ah_l = (const float*)d_in[12];
    const float* W2_ah_r = (const float*)d_in[13];
    const float* W2_ha_l = (const float*)d_in[14];
    const float* W2_ha_r = (const float*)d_in[15];
    const float* b1_ah_l = (const float*)d_in[16];
    const float* b1_ha_l = (const float*)d_in[17];
    const float* b2_ah_l = (const float*)d_in[18];
    const float* b2_ha_l = (const float*)d_in[19];
    const float* W_lin1  = (const float*)d_in[20];
    const float* b_lin1  = (const float*)d_in[21];
    const float* W_lin2  = (const float*)d_in[22];
    const float* b_lin2  = (const float*)d_in[23];

    // workspace layout
    float* ws = (float*)d_ws;
    float* deg_h = ws;                       ws += NH;
    float* deg_a = ws;                       ws += NA;
    float* agg_h = ws;                       ws += (size_t)NH * DD;
    float* agg_a = ws;                       ws += (size_t)NA * DD;
    float* h_h   = ws;                       ws += (size_t)NH * DD;
    float* h_a   = ws;                       ws += (size_t)NA * DD;
    float* z_h   = ws;                       ws += (size_t)NH * DD;
    float* z_a   = ws;                       ws += (size_t)NA * DD;

    // output layout: pred | z1 | z2
    float* pred = (float*)d_out;
    float* z1   = pred + ELN;
    float* z2   = z1 + (size_t)ELN * 2 * DD;

    // ---- degrees (shared by both layers: same edge lists) ----
    hipMemsetAsync(deg_h, 0, sizeof(float) * NH, stream);
    hipMemsetAsync(deg_a, 0, sizeof(float) * NA, stream);
    hipMemsetAsync(agg_h, 0, sizeof(float) * (size_t)NH * DD, stream);
    hipMemsetAsync(agg_a, 0, sizeof(float) * (size_t)NA * DD, stream);

    deg_count_kernel<<<(NE + 255) / 256, 256, 0, stream>>>(ah_dst, deg_h, NE);
    deg_count_kernel<<<(NE + 255) / 256, 256, 0, stream>>>(ha_dst, deg_a, NE);
    rdeg_kernel<<<(NH + 255) / 256, 256, 0, stream>>>(deg_h, NH);
    rdeg_kernel<<<(NA + 255) / 256, 256, 0, stream>>>(deg_a, NA);

    // ---- layer 1: scatter + fused transform (relu) ----
    scatter_kernel<<<(NE + 7) / 8, 256, 0, stream>>>(x_author, ah_src, ah_dst, agg_h, NE);
    scatter_kernel<<<(NE + 7) / 8, 256, 0, stream>>>(x_hotel,  ha_src, ha_dst, agg_a, NE);

    sage_gemm_kernel<<<NH / 16, 256, 0, stream>>>(agg_h, deg_h, W1_ah_l, b1_ah_l,
                                                  x_hotel,  W1_ah_r, h_h, DD, 1);
    sage_gemm_kernel<<<NA / 16, 256, 0, stream>>>(agg_a, deg_a, W1_ha_l, b1_ha_l,
                                                  x_author, W1_ha_r, h_a, DD, 1);

    // ---- layer 2: re-zero agg, scatter h, transform (no relu) ----
    hipMemsetAsync(agg_h, 0, sizeof(float) * (size_t)NH * DD, stream);
    hipMemsetAsync(agg_a, 0, sizeof(float) * (size_t)NA * DD, stream);
    scatter_kernel<<<(NE + 7) / 8, 256, 0, stream>>>(h_a, ah_src, ah_dst, agg_h, NE);
    scatter_kernel<<<(NE + 7) / 8, 256, 0, stream>>>(h_h, ha_src, ha_dst, agg_a, NE);

    sage_gemm_kernel<<<NH / 16, 256, 0, stream>>>(agg_h, deg_h, W2_ah_l, b2_ah_l,
                                                  h_h, W2_ah_r, z_h, DD, 0);
    sage_gemm_kernel<<<NA / 16, 256, 0, stream>>>(agg_a, deg_a, W2_ha_l, b2_ha_l,
                                                  h_a, W2_ha_r, z_a, DD, 0);

    // ---- decoder ----
    concat_kernel<<<(ELN + 7) / 8, 256, 0, stream>>>(z_a, z_h, eli_row, eli_col, z1, ELN);

    // z2 = relu(z1 @ W_lin1 + b_lin1), K = 256, no self term
    sage_gemm_kernel<<<ELN / 16, 256, 0, stream>>>(z1, nullptr, W_lin1, b_lin1,
                                                   nullptr, nullptr, z2, 2 * DD, 1);

    pred_kernel<<<(ELN + 255) / 256, 256, 0, stream>>>(z2, W_lin2, b_lin2, pred, ELN);
}